// SRGNNClassifierV2_43636867727645
// MI455X (gfx1250) — compile-verified
//
#include <hip/hip_runtime.h>
#include <hip/hip_bf16.h>
#include <math.h>

// ---------------------------------------------------------------------------
// Problem constants (from reference)
// ---------------------------------------------------------------------------
#define BGR   2048            // graphs
#define NPG   64              // nodes per graph
#define DFEAT 64              // node feature dim
#define NH    2               // heads
#define DH    32              // per-head dim
#define NLAY  4               // GAT layers
#define EDIM  16              // edge feature dim
#define PERND 8               // edges per node
#define NNODE (BGR * NPG)         // 131072
#define NEDGE (NNODE * PERND)     // 1048576
#define NHS   64
#define NEL   8

typedef __attribute__((ext_vector_type(2))) float v2f;
typedef __attribute__((ext_vector_type(8))) float v8f;
typedef __attribute__((ext_vector_type(4))) unsigned int u32x4;
typedef __attribute__((ext_vector_type(4))) int i32x4;
typedef __attribute__((ext_vector_type(8))) int i32x8;

// ---------------------------------------------------------------------------
// CDNA5 f32 WMMA: D(16x16) = A(16x4) * B(4x16) + C
// A: lanes 0-15 -> M=lane, K={k,k+1}; lanes 16-31 -> K={k+2,k+3}
// B (mirror): lanes 0-15 -> N=lane, rows {k,k+1}; lanes 16-31 -> {k+2,k+3}
// C/D: acc[r] holds row (r + 8*(lane>=16)), col (lane&15)
// ---------------------------------------------------------------------------
__device__ __forceinline__ v8f wmma4_f32(v2f a, v2f b, v8f c) {
  return __builtin_amdgcn_wmma_f32_16x16x4_f32(
      /*neg_a=*/false, a, /*neg_b=*/false, b,
      /*c_mod=*/(short)0, c, /*reuse_a=*/false, /*reuse_b=*/false);
}

// ---------------------------------------------------------------------------
// TDM: DMA a contiguous block of `total` f32 elements from global into LDS.
// D# per ISA ch.8: group0 = {count=1 | lds_addr | global_addr(57b) | type=2},
// group1 = {data_size=4B, 1-D tensor: dim0=total, dim1=1, tile0=total,
//           tile1=1, stride0=total}.  Issued by one wave; TENSORcnt tracks.
// ---------------------------------------------------------------------------
__device__ __forceinline__ void tdm_load_to_lds(const float* gptr,
                                                unsigned lds_off,
                                                unsigned total) {
  const unsigned long long ga = (unsigned long long)(size_t)gptr;
  u32x4 g0;
  g0[0] = 1u;                                            // count=1 (valid)
  g0[1] = lds_off;                                       // lds_addr (bytes)
  g0[2] = (unsigned)(ga & 0xFFFFFFFFu);                  // global_addr[31:0]
  g0[3] = (unsigned)((ga >> 32) & 0x01FFFFFFu) | (2u << 30); // [56:32]|type=2
  i32x8 g1;
  g1[0] = (int)(2u << 16);                               // data_size = 4B
  g1[1] = (int)((total & 0xFFFFu) << 16);                // tensor_dim0[15:0]
  g1[2] = (int)(((total >> 16) & 0xFFFFu) | (1u << 16)); // dim0[31:16],dim1=1
  g1[3] = (int)((total & 0xFFFFu) << 16);                // tile_dim0 = total
  g1[4] = 1;                                             // tile_dim1=1,tile2=0
  g1[5] = (int)total;                                    // dim0_stride[31:0]
  g1[6] = 0;                                             // stride hi / dim1_stride
  g1[7] = 0;
  const i32x4 z4 = {0, 0, 0, 0};
#if __clang_major__ >= 23
  const i32x8 z8 = {0, 0, 0, 0, 0, 0, 0, 0};
  __builtin_amdgcn_tensor_load_to_lds(g0, g1, z4, z4, z8, 0);
#else
  __builtin_amdgcn_tensor_load_to_lds(g0, g1, z4, z4, 0);
#endif
}

// ---------------------------------------------------------------------------
// Persistent fp32 WMMA GEMM with TDM-staged weights:
//   C[M,NOUT] = A[M,K] @ W[K,NOUT] (+bias)
// W (K*NOUT floats, <= 64KB) is DMA'd to LDS once per block; 8 waves then
// grid-stride over 128-row strips reading B fragments from LDS.
// ---------------------------------------------------------------------------
template <int K, int NOUT, bool BIAS>
__global__ __launch_bounds__(256)
void wmma_gemm_lds_kernel(const float* __restrict__ A, const float* __restrict__ W,
                          const float* __restrict__ bias, float* __restrict__ C,
                          int M) {
  __shared__ float Wlds[K * NOUT];
  if (threadIdx.x < 32) {                           // wave 0 issues the DMA
    tdm_load_to_lds(W, (unsigned)(uintptr_t)(void*)&Wlds[0], (unsigned)(K * NOUT));
    __builtin_amdgcn_s_wait_tensorcnt(0);
  }
  __syncthreads();

  const int lane = threadIdx.x & 31;
  const int wave = threadIdx.x >> 5;
  const int half = lane >> 4;                       // 0: lanes 0-15, 1: 16-31
  const int l16  = lane & 15;
  const int nstrips = M / 128;                      // 8 waves x 16 rows

  for (int strip = blockIdx.x; strip < nstrips; strip += gridDim.x) {
    const int m0 = strip * 128 + wave * 16;
    const float* arow = A + (size_t)(m0 + l16) * K;
    for (int n0 = 0; n0 < NOUT; n0 += 16) {
      v8f acc = {0.f, 0.f, 0.f, 0.f, 0.f, 0.f, 0.f, 0.f};
#pragma unroll
      for (int k = 0; k < K; k += 4) {
        const int ka = k + 2 * half;
        v2f a = *(const v2f*)(arow + ka);           // 8B-aligned b64 load
        v2f b;
        b.x = Wlds[ka * NOUT + n0 + l16];
        b.y = Wlds[(ka + 1) * NOUT + n0 + l16];
        acc = wmma4_f32(a, b, acc);
      }
      const float bv = BIAS ? bias[n0 + l16] : 0.0f;
#pragma unroll
      for (int r = 0; r < 8; ++r) {
        const int row = m0 + r + 8 * half;
        C[(size_t)row * NOUT + n0 + l16] = acc[r] + bv;
      }
    }
  }
}

// ---------------------------------------------------------------------------
// Fills
// ---------------------------------------------------------------------------
__global__ void fill_kernel(float* __restrict__ p, float v, size_t n) {
  size_t i = (size_t)blockIdx.x * blockDim.x + threadIdx.x;
  if (i < n) p[i] = v;
}

__global__ void m_fixup_kernel(float* __restrict__ m, size_t n) {
  size_t i = (size_t)blockIdx.x * blockDim.x + threadIdx.x;
  if (i < n) {
    float v = m[i];
    m[i] = isfinite(v) ? v : 0.0f;
  }
}

// ---------------------------------------------------------------------------
// Float atomic max via CAS loop (handles negatives correctly)
// ---------------------------------------------------------------------------
__device__ __forceinline__ void atomicMaxF(float* addr, float val) {
  unsigned int* ua = (unsigned int*)addr;
  unsigned int old = *ua;
  while (true) {
    float oldf = __uint_as_float(old);
    if (oldf >= val) break;
    unsigned int assumed = old;
    old = atomicCAS(ua, assumed, __float_as_uint(val));
    if (old == assumed) break;
  }
}

// ---------------------------------------------------------------------------
// Edge pass 1: s = lrelu(gl[src] + gr[dst] + e@We_l, 0.2); logits = <s,att>
//              (ge recomputed inline: avoids a 268MB/layer intermediate)
//              segment-max into m[dst][h] via atomics.
// ---------------------------------------------------------------------------
__global__ __launch_bounds__(256)
void edge_logits_kernel(const float* __restrict__ gl, const float* __restrict__ gr,
                        const float* __restrict__ eenc, const int* __restrict__ ei,
                        const float* __restrict__ We_l, const float* __restrict__ att_l,
                        float* __restrict__ logits, float* __restrict__ m) {
  const int e = blockIdx.x * blockDim.x + threadIdx.x;
  if (e >= NEDGE) return;
  const int s = ei[e];
  const int d = ei[NEDGE + e];
  float ee[EDIM];
#pragma unroll
  for (int k = 0; k < EDIM; ++k) ee[k] = eenc[(size_t)e * EDIM + k];
  const float* glr = gl + (size_t)s * DFEAT;
  const float* grr = gr + (size_t)d * DFEAT;
#pragma unroll
  for (int h = 0; h < NH; ++h) {
    float lg = 0.0f;
    for (int dd = 0; dd < DH; ++dd) {
      const int j = h * DH + dd;
      float ge = 0.0f;
#pragma unroll
      for (int k = 0; k < EDIM; ++k) ge += ee[k] * We_l[k * (NH * DH) + j];
      float v = glr[j] + grr[j] + ge;
      v = (v > 0.0f) ? v : 0.2f * v;          // leaky_relu(., 0.2)
      lg += v * att_l[j];
    }
    logits[(size_t)e * NH + h] = lg;
    atomicMaxF(&m[(size_t)d * NH + h], lg);
  }
}

// ---------------------------------------------------------------------------
// Edge pass 2: ex = exp(logits - m[dst]); denom[dst] += ex (atomic)
// ---------------------------------------------------------------------------
__global__ __launch_bounds__(256)
void edge_exp_kernel(const int* __restrict__ ei, const float* __restrict__ m,
                     float* __restrict__ logits, float* __restrict__ denom) {
  const int e = blockIdx.x * blockDim.x + threadIdx.x;
  if (e >= NEDGE) return;
  const int d = ei[NEDGE + e];
#pragma unroll
  for (int h = 0; h < NH; ++h) {
    const float ex = expf(logits[(size_t)e * NH + h] - m[(size_t)d * NH + h]);
    logits[(size_t)e * NH + h] = ex;          // overwrite with ex
    atomicAdd(&denom[(size_t)d * NH + h], ex);
  }
}

// ---------------------------------------------------------------------------
// Edge pass 3: agg[dst][j] += (ex/(denom+1e-16)) * gl[src][j]
// One thread per (edge, feature).
// ---------------------------------------------------------------------------
__global__ __launch_bounds__(256)
void edge_aggregate_kernel(const int* __restrict__ ei, const float* __restrict__ ex,
                           const float* __restrict__ denom, const float* __restrict__ gl,
                           float* __restrict__ agg) {
  const size_t idx = (size_t)blockIdx.x * blockDim.x + threadIdx.x;
  if (idx >= (size_t)NEDGE * DFEAT) return;
  const int e = (int)(idx >> 6);
  const int j = (int)(idx & 63);
  const int s = ei[e];
  const int d = ei[NEDGE + e];
  const int h = j >> 5;
  const float a = ex[(size_t)e * NH + h] /
                  (denom[(size_t)d * NH + h] + 1e-16f);
  atomicAdd(&agg[(size_t)d * DFEAT + j], a * gl[(size_t)s * DFEAT + j]);
}

// ---------------------------------------------------------------------------
// Node update: h = lrelu(agg + gat_b, 0.01) + h ; also scatter into hcat
// ---------------------------------------------------------------------------
__global__ __launch_bounds__(256)
void node_update_kernel(float* __restrict__ h, float* __restrict__ hcat,
                        const float* __restrict__ agg, const float* __restrict__ gb,
                        int layer) {
  const size_t idx = (size_t)blockIdx.x * blockDim.x + threadIdx.x;
  if (idx >= (size_t)NNODE * DFEAT) return;
  const int n = (int)(idx >> 6);
  const int j = (int)(idx & 63);
  float v = agg[idx] + gb[j];
  v = (v > 0.0f) ? v : 0.01f * v;             // leaky_relu(., 0.01)
  v += h[idx];
  h[idx] = v;
  hcat[(size_t)n * (NLAY * DFEAT) + layer * DFEAT + j] = v;
}

// ---------------------------------------------------------------------------
// SetRep: t_sum[b][nhs] += sum_over_nodes max_over_8cols relu(out @ Wc)
// WMMA GEMM with fused relu + group-of-8 column max (shfl_xor) + atomic sum.
// K=64 -> A fragments preloaded; loop 32 column tiles of Wc[64,512].
// ---------------------------------------------------------------------------
__global__ __launch_bounds__(256)
void setrep_kernel(const float* __restrict__ out, const float* __restrict__ Wc,
                   float* __restrict__ t_sum) {
  const int lane = threadIdx.x & 31;
  const int wave = threadIdx.x >> 5;
  const int m0 = (blockIdx.x * 8 + wave) * 16;
  if (m0 >= NNODE) return;
  const int half = lane >> 4;
  const int l16  = lane & 15;
  const float* arow = out + (size_t)(m0 + l16) * DFEAT;

  v2f afrag[16];
#pragma unroll
  for (int kk = 0; kk < 16; ++kk)
    afrag[kk] = *(const v2f*)(arow + kk * 4 + 2 * half);

  const int KOUT = NHS * NEL;                 // 512
  for (int n0 = 0; n0 < KOUT; n0 += 16) {
    v8f acc = {0.f, 0.f, 0.f, 0.f, 0.f, 0.f, 0.f, 0.f};
#pragma unroll
    for (int kk = 0; kk < 16; ++kk) {
      const int ka = kk * 4 + 2 * half;
      v2f b;
      const float* bp = Wc + (size_t)ka * KOUT + n0 + l16;
      b.x = bp[0];
      b.y = bp[KOUT];
      acc = wmma4_f32(afrag[kk], b, acc);
    }
    float vals[8];
#pragma unroll
    for (int r = 0; r < 8; ++r) vals[r] = fmaxf(acc[r], 0.0f);
    // max over 8 consecutive columns == 8-lane group (xor masks stay in group)
#pragma unroll
    for (int ofs = 1; ofs < 8; ofs <<= 1)
#pragma unroll
      for (int r = 0; r < 8; ++r)
        vals[r] = fmaxf(vals[r], __shfl_xor(vals[r], ofs, 32));
    if ((lane & 7) == 0) {
      const int nhs = (n0 >> 3) + ((l16 >= 8) ? 1 : 0);
#pragma unroll
      for (int r = 0; r < 8; ++r) {
        const int node = m0 + r + 8 * half;
        atomicAdd(&t_sum[(size_t)(node >> 6) * NHS + nhs], vals[r]);
      }
    }
  }
}

// ---------------------------------------------------------------------------
// Per-graph head: t = relu(t_sum @ fc1_W + fc1_b); p = mean(out over graph);
// logit = [t,p] @ mlp_W + mlp_b.  One block of 64 threads per graph.
// ---------------------------------------------------------------------------
__global__ __launch_bounds__(64)
void head_kernel(const float* __restrict__ t_sum, const float* __restrict__ fc1_W,
                 const float* __restrict__ fc1_b, const float* __restrict__ out,
                 const float* __restrict__ mlp_W, const float* __restrict__ mlp_b,
                 float* __restrict__ logits_out) {
  const int b = blockIdx.x;
  const int j = threadIdx.x;                  // 0..63
  __shared__ float ts[NHS];
  __shared__ float red[64];
  ts[j] = t_sum[(size_t)b * NHS + j];
  __syncthreads();

  float acc = fc1_b[j];
  for (int k = 0; k < NHS; ++k) acc += ts[k] * fc1_W[k * DFEAT + j];
  const float t = fmaxf(acc, 0.0f);

  float p = 0.0f;
  const float* base = out + (size_t)b * NPG * DFEAT;
  for (int n = 0; n < NPG; ++n) p += base[(size_t)n * DFEAT + j];
  p *= (1.0f / (float)NPG);

  red[j] = t * mlp_W[j] + p * mlp_W[DFEAT + j];
  __syncthreads();
  for (int s = 32; s > 0; s >>= 1) {
    if (j < s) red[j] += red[j + s];
    __syncthreads();
  }
  if (j == 0) logits_out[b] = red[0] + mlp_b[0];
}

// ---------------------------------------------------------------------------
// Host launcher
// ---------------------------------------------------------------------------
static inline unsigned cdiv(size_t a, size_t b) { return (unsigned)((a + b - 1) / b); }
static inline unsigned gmin(unsigned a, unsigned b) { return a < b ? a : b; }

extern "C" void kernel_launch(void* const* d_in, const int* in_sizes, int n_in,
                              void* d_out, int out_size, void* d_ws, size_t ws_size,
                              hipStream_t stream) {
  const float* x        = (const float*)d_in[0];
  const float* edgeattr = (const float*)d_in[1];
  const int*   ei       = (const int*)  d_in[2];
  // d_in[3] = batch (implied by layout: node n -> graph n/64)
  const float* Wn       = (const float*)d_in[4];
  const float* bn       = (const float*)d_in[5];
  const float* We       = (const float*)d_in[6];
  const float* be       = (const float*)d_in[7];
  const float* gat_Wl   = (const float*)d_in[8];
  const float* gat_Wr   = (const float*)d_in[9];
  const float* gat_We   = (const float*)d_in[10];
  const float* gat_att  = (const float*)d_in[11];
  const float* gat_b    = (const float*)d_in[12];
  const float* jk_W     = (const float*)d_in[13];
  const float* jk_b     = (const float*)d_in[14];
  const float* Wc       = (const float*)d_in[15];
  const float* fc1_W    = (const float*)d_in[16];
  const float* fc1_b    = (const float*)d_in[17];
  const float* mlp_W    = (const float*)d_in[18];
  const float* mlp_b    = (const float*)d_in[19];
  float* logits_out     = (float*)d_out;

  // -------- workspace partition (floats) --------
  float* ws = (float*)d_ws;
  float* h      = ws;                       ws += (size_t)NNODE * DFEAT;        // 8.4M
  float* hcat   = ws;                       ws += (size_t)NNODE * NLAY * DFEAT; // 33.6M
  float* eenc   = ws;                       ws += (size_t)NEDGE * EDIM;         // 16.8M
  float* gl     = ws;                       ws += (size_t)NNODE * DFEAT;
  float* gr     = ws;                       ws += (size_t)NNODE * DFEAT;
  float* elog   = ws;                       ws += (size_t)NEDGE * NH;           // 2.1M
  float* mmax   = ws;                       ws += (size_t)NNODE * NH;
  float* denom  = ws;                       ws += (size_t)NNODE * NH;
  float* agg    = ws;                       ws += (size_t)NNODE * DFEAT;
  float* outb   = ws;                       ws += (size_t)NNODE * DFEAT;
  float* tsum   = ws;                       ws += (size_t)BGR * NHS;

  const dim3 blk256(256);
  const unsigned gNode = gmin(NNODE / 128, 1024);   // persistent grid
  const unsigned gEdge = gmin(NEDGE / 128, 1024);

  // 1) node encoder: h = x @ Wn + bn
  wmma_gemm_lds_kernel<DFEAT, DFEAT, true>
      <<<gNode, blk256, 0, stream>>>(x, Wn, bn, h, NNODE);
  // 2) edge encoder: eenc = edge_attr @ We + be
  wmma_gemm_lds_kernel<EDIM, EDIM, true>
      <<<gEdge, blk256, 0, stream>>>(edgeattr, We, be, eenc, NEDGE);

  // 3) GATv2 stack
  for (int l = 0; l < NLAY; ++l) {
    const float* Wl_l  = gat_Wl  + (size_t)l * DFEAT * DFEAT;
    const float* Wr_l  = gat_Wr  + (size_t)l * DFEAT * DFEAT;
    const float* We_l  = gat_We  + (size_t)l * EDIM * (NH * DH);
    const float* att_l = gat_att + (size_t)l * (NH * DH);
    const float* gb_l  = gat_b   + (size_t)l * (NH * DH);

    wmma_gemm_lds_kernel<DFEAT, DFEAT, false>
        <<<gNode, blk256, 0, stream>>>(h, Wl_l, nullptr, gl, NNODE);
    wmma_gemm_lds_kernel<DFEAT, DFEAT, false>
        <<<gNode, blk256, 0, stream>>>(h, Wr_l, nullptr, gr, NNODE);

    fill_kernel<<<cdiv((size_t)NNODE * NH, 256), blk256, 0, stream>>>(mmax, -INFINITY, (size_t)NNODE * NH);
    fill_kernel<<<cdiv((size_t)NNODE * NH, 256), blk256, 0, stream>>>(denom, 0.0f, (size_t)NNODE * NH);
    fill_kernel<<<cdiv((size_t)NNODE * DFEAT, 256), blk256, 0, stream>>>(agg, 0.0f, (size_t)NNODE * DFEAT);

    edge_logits_kernel<<<NEDGE / 256, blk256, 0, stream>>>(gl, gr, eenc, ei, We_l, att_l, elog, mmax);
    m_fixup_kernel<<<cdiv((size_t)NNODE * NH, 256), blk256, 0, stream>>>(mmax, (size_t)NNODE * NH);
    edge_exp_kernel<<<NEDGE / 256, blk256, 0, stream>>>(ei, mmax, elog, denom);
    edge_aggregate_kernel<<<cdiv((size_t)NEDGE * DFEAT, 256), blk256, 0, stream>>>(ei, elog, denom, gl, agg);
    node_update_kernel<<<cdiv((size_t)NNODE * DFEAT, 256), blk256, 0, stream>>>(h, hcat, agg, gb_l, l);
  }

  // 4) JK: out = hcat[N,256] @ jk_W + jk_b   (64KB weight staged via TDM)
  wmma_gemm_lds_kernel<NLAY * DFEAT, DFEAT, true>
      <<<gNode, blk256, 0, stream>>>(hcat, jk_W, jk_b, outb, NNODE);

  // 5) SetRep: t_sum = sum_n max_8 relu(out @ Wc)
  fill_kernel<<<cdiv((size_t)BGR * NHS, 256), blk256, 0, stream>>>(tsum, 0.0f, (size_t)BGR * NHS);
  setrep_kernel<<<NNODE / 128, blk256, 0, stream>>>(outb, Wc, tsum);

  // 6) head: logits = [relu(t_sum@fc1+b), mean(out)] @ mlp_W + mlp_b
  head_kernel<<<BGR, dim3(64), 0, stream>>>(tsum, fc1_W, fc1_b, outb, mlp_W, mlp_b, logits_out);
}